// HMQSoftmax_59983513256183
// MI455X (gfx1250) — compile-verified
//
#include <hip/hip_runtime.h>
#include <stdint.h>

// HMQ softmax + BFP quant, MI455X (gfx1250, wave32).
// Bandwidth-bound: 1.07 GB traffic -> ~46us floor @ 23.3 TB/s.
// Round 3: shrink static VALU below the HBM roofline crossover.
//  - <2 x float> ext-vectors -> V_PK_MUL_F32 / V_PK_ADD_F32 (VOP3P packed f32)
//  - quant via magic-number RNE round: min((r+C)-C, 127*2^sh), C = 1.5*2^(23+sh)
//  - balanced max tree (v_max3-friendly), split accumulators (VOPD pairing)
// Identities kept from round 2: e = 2^q exact; bf16(bf16(e*s)*s) = e*bf16(s*s).
// CDNA5 paths: GLOBAL_LOAD_ASYNC_TO_LDS_B128 (ASYNCcnt), s_wait_asynccnt,
// split barriers, wave32 shuffle reduction, b128 LDS/VMEM.

typedef float f32x2 __attribute__((ext_vector_type(2)));

#define ROW_LEN 2048
#define TPB     128   // 4 waves; each thread owns one 16-elem BFP block

// round f32 -> nearest-even bf16, returned as f32 (row scalars only)
__device__ __forceinline__ float bf16r(float f) {
    uint32_t u = __float_as_uint(f);
    u = (u + 0x7FFFu + ((u >> 16) & 1u)) & 0xFFFF0000u;
    return __uint_as_float(u);
}

__global__ __launch_bounds__(TPB) void hmq_softmax_bfp_kernel(
    const float* __restrict__ x, float* __restrict__ out)
{
    const int row = blockIdx.x;
    const int t   = threadIdx.x;

    __shared__ __align__(16) float smem[ROW_LEN];
    __shared__ float wsum[TPB / 32];

    // ---------------- async global -> LDS row staging (ASYNCcnt path) -------
    const float*   g       = x + (size_t)row * ROW_LEN;
    const uint32_t ldsBase = (uint32_t)(uintptr_t)&smem[0];  // AS(3) offset
    #pragma unroll
    for (int j = 0; j < 4; ++j) {
        const int      fidx = (t + j * TPB) * 4;             // 16B per lane
        const uint32_t lds  = ldsBase + (uint32_t)fidx * 4u;
        const uint64_t ga   = (uint64_t)(uintptr_t)(g + fidx);
        asm volatile("global_load_async_to_lds_b128 %0, %1, off"
                     :: "v"(lds), "v"(ga) : "memory");
    }
    asm volatile("s_wait_asynccnt 0" ::: "memory");
    __syncthreads();

    // ---------------- pass 1: e = 2^floor(x/ln2) (exact) + local sum --------
    float e[16];
    f32x2 accA = {0.0f, 0.0f}, accB = {0.0f, 0.0f};
    const f32x2 k2 = {1.44269504088896340736f, 1.44269504088896340736f};
    const float4* s4 = (const float4*)smem;
    #pragma unroll
    for (int j = 0; j < 4; ++j) {
        float4 v = s4[t * 4 + j];                  // ds_load_b128
        f32x2 lo = {v.x, v.y};
        f32x2 hi = {v.z, v.w};
        lo = lo * k2;                              // v_pk_mul_f32
        hi = hi * k2;
        const int q0 = __float2int_rd(lo.x), q1 = __float2int_rd(lo.y);
        const int q2 = __float2int_rd(hi.x), q3 = __float2int_rd(hi.y);
        f32x2 e01 = {ldexpf(1.0f, q0), ldexpf(1.0f, q1)};   // exact 2^q
        f32x2 e23 = {ldexpf(1.0f, q2), ldexpf(1.0f, q3)};
        e[4*j+0] = e01.x; e[4*j+1] = e01.y;
        e[4*j+2] = e23.x; e[4*j+3] = e23.y;
        accA += e01;                               // v_pk_add_f32
        accB += e23;
    }
    const f32x2 accs = accA + accB;
    float local = accs.x + accs.y;

    // ---------------- row-sum reduction (wave32 shfl + LDS) -----------------
    #pragma unroll
    for (int off = 16; off > 0; off >>= 1)
        local += __shfl_xor(local, off, 32);
    const int lane = t & 31, wave = t >> 5;
    if (lane == 0) wsum[wave] = local;
    __syncthreads();
    const float total = wsum[0] + wsum[1] + wsum[2] + wsum[3];

    // ---------------- Quake isqrt on bf16 bit pattern (row scalar) ----------
    const float    S  = bf16r(total);
    const uint32_t Sb = __float_as_uint(S) >> 16;            // bf16 bits
    const int16_t  ii = (int16_t)(24375 - (int16_t)((int16_t)Sb >> 1));
    const float    y0 = __uint_as_float(((uint32_t)(uint16_t)ii) << 16);
    const float y_tmp   = bf16r(y0 * y0);
    const float x_tmp   = bf16r(S * 0.5f);
    const float mul_tmp = bf16r(x_tmp * y_tmp);
    const float sub_tmp = bf16r(1.5f - mul_tmp);
    const float s       = bf16r(y0 * sub_tmp);               // 1/sqrt(sum), bf16
    const float ss      = bf16r(s * s);   // bf16(bf16(e*s)*s) == e * ss

    // ---------------- normalize (pk mul) + balanced block max ---------------
    float r[16];
    const f32x2 ssv = {ss, ss};
    #pragma unroll
    for (int i = 0; i < 16; i += 2) {
        f32x2 ri = {e[i], e[i + 1]};
        ri = ri * ssv;                             // v_pk_mul_f32; exact bf16 val
        r[i] = ri.x; r[i + 1] = ri.y;              // r >= 0
    }
    float m[8];
    #pragma unroll
    for (int i = 0; i < 8; ++i) m[i] = fmaxf(r[2*i], r[2*i+1]);
    #pragma unroll
    for (int i = 0; i < 4; ++i) m[i] = fmaxf(m[i], m[i+4]);
    const float maxabs = fmaxf(fmaxf(m[0], m[1]), fmaxf(m[2], m[3]));

    // ---------------- BFP quant: shared exponent, 8-bit mantissa ------------
    const float safe = fmaxf(maxabs, 1e-38f);
    // floor(log2(safe)) via exponent bits; exact for safe >= 1e-38 incl. the
    // [2^-127, 2^-126) denormal bin the clamp lands in.
    const int E  = (int)((__float_as_uint(safe) >> 23) & 0xFFu) - 127;
    const int sh = E - 7;                         // lsb exponent of mantissa
    const float C = ldexpf(12582912.0f, sh);      // 1.5*2^23 * 2^sh (magic RNE)
    const float B = ldexpf(127.0f, sh);           // clamp bound, pre-scaled
    const f32x2 Cv = {C, C};

    float4* outv = (float4*)(out + (size_t)row * ROW_LEN);
    #pragma unroll
    for (int j = 0; j < 4; ++j) {
        f32x2 a = {r[4*j+0], r[4*j+1]};
        f32x2 b = {r[4*j+2], r[4*j+3]};
        a = (a + Cv) - Cv;                        // pk add/sub: RNE to 2^sh grid
        b = (b + Cv) - Cv;                        // (0 <= r < 2^(sh+8) << 2^(sh+22))
        float4 o4;
        o4.x = fminf(a.x, B);                     // == min(round(r/2^sh),127)*2^sh
        o4.y = fminf(a.y, B);
        o4.z = fminf(b.x, B);
        o4.w = fminf(b.y, B);
        outv[t * 4 + j] = o4;                     // global_store_b128
    }
}

extern "C" void kernel_launch(void* const* d_in, const int* in_sizes, int n_in,
                              void* d_out, int out_size, void* d_ws, size_t ws_size,
                              hipStream_t stream) {
    const float* x   = (const float*)d_in[0];
    float*       out = (float*)d_out;
    const int n    = in_sizes[0];
    const int rows = n / ROW_LEN;                 // 65536 for ref shape
    hmq_softmax_bfp_kernel<<<rows, TPB, 0, stream>>>(x, out);
}